// polarTaskAlignedAssigner_90915867722336
// MI455X (gfx1250) — compile-verified
//
#include <hip/hip_runtime.h>
#include <hip/hip_bf16.h>
#include <math.h>

#define BS   4
#define NM   4
#define NA   8400
#define NC   80
#define RAYS 36
#define NPTS 360

// Flat float offsets of each output in d_out (reference return order).
#define OFF_LABELS 0
#define OFF_BBOXES (BS*NA)                        // 33600
#define OFF_SCORES (OFF_BBOXES + BS*NA*4)         // 168000
#define OFF_MASK   (OFF_SCORES + BS*NA*NC)        // 2856000
#define OFF_TGI    (OFF_MASK + BS*NM*NA)          // 2990400
#define OFF_GTD    (OFF_TGI + BS*NA)              // 3024000
#define OFF_CNESS  (OFF_GTD + (size_t)BS*NM*NA*RAYS) // 7862400
#define OFF_FG     (OFF_CNESS + BS*NM*NA)         // 7996800

// ---------------- Kernel A: assignment / targets (1 thread per (b,anchor)) ---
__global__ void __launch_bounds__(256)
assign_kernel(const float* __restrict__ pd_bboxes,
              const float* __restrict__ anc,
              const int*   __restrict__ gt_labels,
              const float* __restrict__ gt_bboxes,
              const float* __restrict__ mask_gt,
              const float* __restrict__ stride_t,
              float* __restrict__ out)
{
    int idx = blockIdx.x * blockDim.x + threadIdx.x;
    if (idx >= BS * NA) return;
    int b = idx / NA, a = idx % NA;

    float ax = anc[2*a], ay = anc[2*a + 1];
    float s15 = stride_t[a] * 1.5f;
    float lo, hi;
    if (a < 6400)      { lo = -1.f;  hi = 64.f;  }
    else if (a < 8000) { lo = 64.f;  hi = 128.f; }
    else               { lo = 128.f; hi = 1e8f;  }

    const float4 pb = ((const float4*)pd_bboxes)[idx];
    float ap = (pb.z - pb.x) * (pb.w - pb.y);

    float pre[NM], ins[NM], ov[NM];
#pragma unroll
    for (int m = 0; m < NM; m++) {
        const float* g = gt_bboxes + (b*NM + m)*4;
        float gx1 = g[0], gy1 = g[1], gx2 = g[2], gy2 = g[3];
        // stride-range gate on max regression distance
        float l = ax - gx1, t = ay - gy1, r = gx2 - ax, btm = gy2 - ay;
        float mrd = fmaxf(fmaxf(l, t), fmaxf(r, btm));
        ins[m] = (mrd >= lo && mrd <= hi) ? 1.f : 0.f;
        // center-radius gate
        float cx = (gx1 + gx2) * 0.5f, cy = (gy1 + gy2) * 0.5f;
        float x1 = fmaxf(cx - s15, gx1), y1 = fmaxf(cy - s15, gy1);
        float x2 = fminf(cx + s15, gx2), y2 = fminf(cy + s15, gy2);
        float cb = fminf(fminf(ax - x1, ay - y1), fminf(x2 - ax, y2 - ay));
        float p = (cb > 0.f) ? mask_gt[b*NM + m] : 0.f;
        pre[m] = p;
        // IoU vs predicted box (masked)
        float ltx = fmaxf(gx1, pb.x), lty = fmaxf(gy1, pb.y);
        float rbx = fminf(gx2, pb.z), rby = fminf(gy2, pb.w);
        float wi = fmaxf(rbx - ltx, 0.f), he = fmaxf(rby - lty, 0.f);
        float inter = wi * he;
        float ag = (gx2 - gx1) * (gy2 - gy1);
        ov[m] = inter / (ag + ap - inter + 1e-9f) * p;
    }

    float fgpre = pre[0] + pre[1] + pre[2] + pre[3];
    int bi = 0; float bo = ov[0];
#pragma unroll
    for (int m = 1; m < NM; m++) if (ov[m] > bo) { bo = ov[m]; bi = m; }  // first-max
    bool multi = fgpre > 1.f;

    float m2[NM];
#pragma unroll
    for (int m = 0; m < NM; m++) m2[m] = multi ? ((m == bi) ? 1.f : 0.f) : pre[m];

    int tgi = 0; bool found = false;
#pragma unroll
    for (int m = 0; m < NM; m++) if (!found && m2[m] > 0.f) { tgi = m; found = true; }

    float fgF = 0.f;
#pragma unroll
    for (int m = 0; m < NM; m++) {
        float mf = m2[m] * ins[m];
        out[OFF_MASK + (size_t)(b*NM + m)*NA + a] = mf;  // mask_pos > 0 (0/1)
        fgF += mf;
    }

    int lab = gt_labels[b*NM + tgi];
    out[OFF_LABELS + idx] = (float)lab;
    const float* tb = gt_bboxes + (b*NM + tgi)*4;
    ((float4*)(out + OFF_BBOXES))[idx] = make_float4(tb[0], tb[1], tb[2], tb[3]);
    float fgflag = (fgF > 0.f) ? 1.f : 0.f;

    // one-hot scores: 20 x b128 stores instead of 80 scalar stores
    float4* sc4 = (float4*)(out + OFF_SCORES) + (size_t)idx * (NC / 4);
#pragma unroll
    for (int j = 0; j < NC / 4; j++) {
        int c0 = j * 4;
        float4 v = make_float4((lab == c0    ) ? fgflag : 0.f,
                               (lab == c0 + 1) ? fgflag : 0.f,
                               (lab == c0 + 2) ? fgflag : 0.f,
                               (lab == c0 + 3) ? fgflag : 0.f);
        sc4[j] = v;
    }
    out[OFF_TGI + idx] = (float)tgi;
    out[OFF_FG  + idx] = fgflag;
}

// ------------- Kernel B: polar gt_dist + centerness (1 block per cell) ------
__global__ void __launch_bounds__(64)
polar_kernel(const float* __restrict__ gt_coor,
             const float* __restrict__ anc,
             float* __restrict__ out)
{
    __shared__ float sh_gxy[NPTS * 2];   // interleaved x,y staged via async DMA
    __shared__ float sh_ang[NPTS];
    __shared__ float sh_dist[NPTS];
    __shared__ float sh_ray[RAYS];

    int cell = blockIdx.x;               // linear (b*NM+m)*NA + a
    int a  = cell % NA;
    int bm = cell / NA;
    int tid = threadIdx.x;

    float mask = out[OFF_MASK + cell];   // written by assign_kernel, same stream
    if (mask <= 0.f) {                   // vast majority: constant fill + exit
        // 36 floats = 9 x float4 per cell, 16B aligned (144B stride)
        if (tid < RAYS / 4)
            ((float4*)(out + OFF_GTD))[(size_t)cell * (RAYS / 4) + tid] =
                make_float4(1e-6f, 1e-6f, 1e-6f, 1e-6f);
        if (tid == 0) out[OFF_CNESS + cell] = 0.f;
        return;
    }

    // --- Stage 720 floats (180 x 16B) of this GT's polygon into LDS using the
    // --- gfx1250 async global->LDS path (ASYNCcnt-tracked, bypasses VGPRs).
    const float* gbase = gt_coor + (size_t)bm * (NPTS * 2);
    uint32_t lds0 = (uint32_t)(uintptr_t)(&sh_gxy[0]); // low 32b of flat ptr = LDS byte offset
    for (int i = tid; i < (NPTS * 2) / 4; i += blockDim.x) {
        uint32_t go = (uint32_t)(i * 16);
        asm volatile("global_load_async_to_lds_b128 %0, %1, %2"
                     :: "v"(lds0 + go), "v"(go), "s"(gbase) : "memory");
    }
    asm volatile("s_wait_asynccnt 0" ::: "memory");
    __syncthreads();

    float ax = anc[2*a], ay = anc[2*a + 1];
    const float RAD2DEG = 57.29577951308232f;
    for (int i = tid; i < NPTS; i += blockDim.x) {
        float dx = sh_gxy[2*i]     - ax;
        float dy = sh_gxy[2*i + 1] - ay;
        float ang = atan2f(dy, dx) * RAD2DEG;
        ang = fmodf(ang, 360.f); if (ang < 0.f) ang += 360.f;   // python % semantics
        sh_ang[i]  = ang;
        sh_dist[i] = sqrtf(dx*dx + dy*dy);
    }
    __syncthreads();

    if (tid < RAYS) {
        float tv = (float)(tid * 10);
        // top-4 smallest angular deviation (stable: earlier index wins ties)
        float d0 = 1e30f, d1 = 1e30f, d2 = 1e30f, d3 = 1e30f;
        float s0 = 0.f, s1 = 0.f, s2 = 0.f, s3 = 0.f;
        for (int i = 0; i < NPTS; i++) {
            float d = fabsf(sh_ang[i] - tv);
            d = (d > 180.f) ? (360.f - d) : d;
            float ds = sh_dist[i];
            if (d < d3) {
                if (d < d2) {
                    d3 = d2; s3 = s2;
                    if (d < d1) {
                        d2 = d1; s2 = s1;
                        if (d < d0) { d1 = d0; s1 = s0; d0 = d; s0 = ds; }
                        else        { d1 = d;  s1 = ds; }
                    } else { d2 = d; s2 = ds; }
                } else { d3 = d; s3 = ds; }
            }
        }
        float v = (d0 > 3.f) ? 1e-6f : fmaxf(fmaxf(s0, s1), fmaxf(s2, s3));
        v = fmaxf(v, 1e-6f);                       // clip
        out[OFF_GTD + (size_t)cell * RAYS + tid] = v;
        sh_ray[tid] = v;
    }
    __syncthreads();

    if (tid == 0) {
        float mn = sh_ray[0], mx = sh_ray[0];
#pragma unroll
        for (int t = 1; t < RAYS; t++) { mn = fminf(mn, sh_ray[t]); mx = fmaxf(mx, sh_ray[t]); }
        out[OFF_CNESS + cell] = sqrtf(mn / mx) * mask;
    }
}

extern "C" void kernel_launch(void* const* d_in, const int* in_sizes, int n_in,
                              void* d_out, int out_size, void* d_ws, size_t ws_size,
                              hipStream_t stream) {
    (void)in_sizes; (void)n_in; (void)out_size; (void)d_ws; (void)ws_size;
    // setup_inputs order: pd_scores(unused), pd_bboxes, anc_points, gt_labels,
    // gt_bboxes, mask_gt, gt_coor, stride_tensor, gt_center(unused),
    // oriimg(unused), h, w (compile-time constants here).
    const float* pd_bboxes = (const float*)d_in[1];
    const float* anc       = (const float*)d_in[2];
    const int*   gt_labels = (const int*)  d_in[3];
    const float* gt_bboxes = (const float*)d_in[4];
    const float* mask_gt   = (const float*)d_in[5];
    const float* gt_coor   = (const float*)d_in[6];
    const float* stride_t  = (const float*)d_in[7];
    float* out = (float*)d_out;

    int nthreads = BS * NA;
    assign_kernel<<<(nthreads + 255) / 256, 256, 0, stream>>>(
        pd_bboxes, anc, gt_labels, gt_bboxes, mask_gt, stride_t, out);

    polar_kernel<<<BS * NM * NA, 64, 0, stream>>>(gt_coor, anc, out);
}